// LearnableGaussianField_88064009437939
// MI455X (gfx1250) — compile-verified
//
#include <hip/hip_runtime.h>
#include <hip/hip_bf16.h>

typedef __attribute__((ext_vector_type(2))) float v2f;
typedef __attribute__((ext_vector_type(8))) float v8f;

// ---------------------------------------------------------------------------
// Kernel 1: per-gaussian preprocessing.
// cov = L L^T + 1e-6 I  (L rebuilt from cov_tril as in the reference),
// A = cov^{-1} (analytic symmetric 3x3 inverse), then emit 12 coefficients
// g[n][k] so that  mahal(b,n) = f(b) . g(n)  with
//   f = [px^2, py^2, pz^2, px*py, px*pz, py*pz, px, py, pz, 1, 0, 0]
//   g = [A00, A11, A22, 2A01, 2A02, 2A12, -2(Am)0, -2(Am)1, -2(Am)2, m^T A m, 0, 0]
// Layout: gP[n*12 + k]  (48-byte stride -> every float2 we read is 8B aligned)
// ---------------------------------------------------------------------------
__global__ void lgf_prep_kernel(const float* __restrict__ means,
                                const float* __restrict__ cov_tril,
                                float* __restrict__ gP, int N) {
  int n = blockIdx.x * blockDim.x + threadIdx.x;
  if (n >= N) return;

  float t0 = cov_tril[n * 6 + 0];
  float t1 = cov_tril[n * 6 + 1];
  float t2 = cov_tril[n * 6 + 2];
  float t3 = cov_tril[n * 6 + 3];
  float t4 = cov_tril[n * 6 + 4];
  float t5 = cov_tril[n * 6 + 5];

  float l00 = __expf(t0);
  float l10 = t1;
  float l11 = __expf(t2);
  float l20 = t3;
  float l21 = t4;
  float l22 = __expf(t5);

  const float eps = 1e-6f;
  float c00 = l00 * l00 + eps;
  float c01 = l10 * l00;
  float c02 = l20 * l00;
  float c11 = l10 * l10 + l11 * l11 + eps;
  float c12 = l20 * l10 + l21 * l11;
  float c22 = l20 * l20 + l21 * l21 + l22 * l22 + eps;

  // Symmetric 3x3 inverse via adjugate.
  float M00 = c11 * c22 - c12 * c12;
  float M01 = c02 * c12 - c01 * c22;
  float M02 = c01 * c12 - c02 * c11;
  float det = c00 * M00 + c01 * M01 + c02 * M02;
  float inv = 1.0f / det;

  float A00 = M00 * inv;
  float A01 = M01 * inv;
  float A02 = M02 * inv;
  float A11 = (c00 * c22 - c02 * c02) * inv;
  float A12 = (c02 * c01 - c00 * c12) * inv;
  float A22 = (c00 * c11 - c01 * c01) * inv;

  float mx = means[n * 3 + 0];
  float my = means[n * 3 + 1];
  float mz = means[n * 3 + 2];

  float Am0 = A00 * mx + A01 * my + A02 * mz;
  float Am1 = A01 * mx + A11 * my + A12 * mz;
  float Am2 = A02 * mx + A12 * my + A22 * mz;
  float kc  = mx * Am0 + my * Am1 + mz * Am2;

  float* g = gP + (size_t)n * 12;
  g[0]  = A00;
  g[1]  = A11;
  g[2]  = A22;
  g[3]  = 2.0f * A01;
  g[4]  = 2.0f * A02;
  g[5]  = 2.0f * A12;
  g[6]  = -2.0f * Am0;
  g[7]  = -2.0f * Am1;
  g[8]  = -2.0f * Am2;
  g[9]  = kc;
  g[10] = 0.0f;   // K padding rows (K=10..11) so K=12 = 3 x WMMA K=4
  g[11] = 0.0f;
}

// ---------------------------------------------------------------------------
// Kernel 2: main evaluation. One wave32 owns a 16-point row block and loops
// over all gaussian 16-column tiles. mahal tile computed with 3 chained
// V_WMMA_F32_16X16X4_F32 (K = 12), then exp + weighted accumulate per lane,
// 16-lane butterfly reduce at the end.
//
// A-matrix 16x4 f32 layout (ISA 7.12.2): lanes 0-15 = rows M, VGPR0=K0,
// VGPR1=K1; lanes 16-31 = same rows, K2/K3.  B-matrix 4x16 mirrors this.
// C/D: VGPR r -> (M=r, N=lane) for lanes 0-15, (M=r+8, N=lane-16) for 16-31.
// ---------------------------------------------------------------------------
__global__ __launch_bounds__(256) void lgf_main_kernel(
    const float* __restrict__ x, const float* __restrict__ gP,
    const float* __restrict__ weights, float* __restrict__ out,
    int batch, int ntiles) {
  const int lane = threadIdx.x & 31;
  const int wave = threadIdx.x >> 5;
  const int rowBlock = blockIdx.x * 8 + wave;   // 8 waves per 256-thread block
  const int rowBase = rowBlock * 16;
  if (rowBase >= batch) return;                 // wave-uniform

  const int col  = lane & 15;
  const int half = lane >> 4;

  // Features of the point this lane mirrors (same for both lane halves).
  const float* p = x + (size_t)(rowBase + col) * 3;
  float px = p[0], py = p[1], pz = p[2];
  float f0 = px * px, f1 = py * py, f2 = pz * pz;
  float f3 = px * py, f4 = px * pz, f5 = py * pz;

  // A operands: lane half selects K offset {0,1} vs {2,3} within each chunk.
  v2f a0, a1, a2;
  a0.x = half ? f2 : f0;    a0.y = half ? f3 : f1;      // K rows 0..3
  a1.x = half ? px : f4;    a1.y = half ? py : f5;      // K rows 4..7
  a2.x = half ? 0.0f : pz;  a2.y = half ? 0.0f : 1.0f;  // K rows 8..11

  float acc[8];
#pragma unroll
  for (int r = 0; r < 8; ++r) acc[r] = 0.0f;

  const int kb = 2 * half;   // per-lane K base within a chunk

  for (int t = 0; t < ntiles; ++t) {
    int n = t * 16 + col;
    const float* gb = gP + (size_t)n * 12 + kb;
    v2f b0 = *(const v2f*)(gb + 0);   // rows kb, kb+1
    v2f b1 = *(const v2f*)(gb + 4);   // rows 4+kb, 5+kb
    v2f b2 = *(const v2f*)(gb + 8);   // rows 8+kb, 9+kb
    float w = weights[n];

    v8f c = {};
    c = __builtin_amdgcn_wmma_f32_16x16x4_f32(false, a0, false, b0,
                                              (short)0, c, false, false);
    c = __builtin_amdgcn_wmma_f32_16x16x4_f32(false, a1, false, b1,
                                              (short)0, c, false, false);
    c = __builtin_amdgcn_wmma_f32_16x16x4_f32(false, a2, false, b2,
                                              (short)0, c, false, false);

#pragma unroll
    for (int r = 0; r < 8; ++r)
      acc[r] = fmaf(w, __expf(-0.5f * c[r]), acc[r]);
  }

  // Reduce over the 16 lanes of each half (xor over bits 0..3 stays in-half).
#pragma unroll
  for (int r = 0; r < 8; ++r) {
    float v = acc[r];
    v += __shfl_xor(v, 1, 32);
    v += __shfl_xor(v, 2, 32);
    v += __shfl_xor(v, 4, 32);
    v += __shfl_xor(v, 8, 32);
    acc[r] = v;
  }

  if (col == 0) {
    // lane 0 -> rows M=0..7, lane 16 -> rows M=8..15
#pragma unroll
    for (int r = 0; r < 8; ++r) out[rowBase + half * 8 + r] = acc[r];
  }
}

// ---------------------------------------------------------------------------
// Launch: inputs are (x[B,3], means[N,3], cov_tril[N,6], weights[N]); output
// is float[B]. d_ws holds the 12xN coefficient matrix (N*12*4 = 96 KB).
// ---------------------------------------------------------------------------
extern "C" void kernel_launch(void* const* d_in, const int* in_sizes, int n_in,
                              void* d_out, int out_size, void* d_ws, size_t ws_size,
                              hipStream_t stream) {
  const float* x        = (const float*)d_in[0];
  const float* means    = (const float*)d_in[1];
  const float* cov_tril = (const float*)d_in[2];
  const float* weights  = (const float*)d_in[3];
  float* out = (float*)d_out;

  int batch = in_sizes[0] / 3;   // 16384
  int N     = in_sizes[3];       // 2048
  float* gP = (float*)d_ws;      // N * 12 floats

  lgf_prep_kernel<<<(N + 255) / 256, 256, 0, stream>>>(means, cov_tril, gP, N);

  int rowBlocks = (batch + 15) / 16;        // 1024
  int blocks    = (rowBlocks + 7) / 8;      // 128 (8 waves / block)
  lgf_main_kernel<<<blocks, 256, 0, stream>>>(x, gP, weights, out,
                                              batch, N / 16);
}